// LDARBCVBBVI_27127013441904
// MI455X (gfx1250) — compile-verified
//
#include <hip/hip_runtime.h>
#include <hip/hip_bf16.h>
#include <stdint.h>

#define V  50257
#define K  64
#define D  512
#define S  4
#define EPS 1e-10f

typedef __attribute__((ext_vector_type(2))) float v2f;
typedef __attribute__((ext_vector_type(8))) float v8f;
typedef int i32x4 __attribute__((__vector_size__(16)));

#define AS1 __attribute__((address_space(1)))
#define AS3 __attribute__((address_space(3)))

#if __has_builtin(__builtin_amdgcn_global_load_async_to_lds_b128) && \
    __has_builtin(__builtin_amdgcn_global_load_async_to_lds_b32)  && \
    __has_builtin(__builtin_amdgcn_s_wait_asynccnt)
#define USE_ASYNC_LDS 1
#else
#define USE_ASYNC_LDS 0
#endif

// ---------------- deterministic counter-based RNG ----------------
__device__ __forceinline__ uint32_t mix64(uint64_t x) {
  x ^= x >> 33; x *= 0xff51afd7ed558ccdULL;
  x ^= x >> 33; x *= 0xc4ceb9fe1a85ec53ULL;
  x ^= x >> 33;
  return (uint32_t)x;
}
__device__ __forceinline__ float u01(uint64_t id, int strm) {
  uint32_t h = mix64(((id << 7) + (uint64_t)strm) ^ 0x9E3779B97F4A7C15ULL);
  return ((float)(h >> 8) + 0.5f) * 5.9604645e-08f;   // (0,1)
}

// Marsaglia-Tsang gamma sampler (shape boost for a<1), bounded rejection
__device__ float gamma_sample(float a, uint64_t id) {
  float boost = 1.0f, ash = a;
  if (a < 1.0f) {
    float u = u01(id, 120);
    boost = __expf(__logf(u) / a);   // may underflow to 0 for tiny a (expected)
    ash = a + 1.0f;
  }
  float dd = ash - (1.0f / 3.0f);
  float cc = rsqrtf(9.0f * dd);
  float x = 0.0f, vv = 1.0f;
  #pragma unroll 1
  for (int i = 0; i < 12; ++i) {
    float u1 = u01(id, 2 * i), u2 = u01(id, 2 * i + 1);
    float rr = sqrtf(fmaxf(-2.0f * __logf(u1), 0.0f));
    x = rr * __cosf(6.2831855f * u2);
    float t = 1.0f + cc * x;
    if (t <= 0.0f) continue;
    vv = t * t * t;
    float u = u01(id, 64 + i);
    if (__logf(u) < 0.5f * x * x + dd - dd * vv + dd * __logf(vv)) break;
  }
  return boost * dd * vv;
}

// ---------------- lambda stats: lamsum[k], sum lgamma(lam) ----------------
__global__ void lam_stats_kernel(const float* __restrict__ tlv,
                                 float* __restrict__ lamsum,
                                 float* __restrict__ lgl) {
  int v = blockIdx.x * 256 + threadIdx.x;
  int k = blockIdx.y;
  float lam = 0.0f, lg = 0.0f;
  if (v < V) {
    lam = __expf(tlv[(size_t)k * V + v]);
    lg  = lgammaf(lam);
  }
  __shared__ float r0[256], r1[256];
  r0[threadIdx.x] = lam; r1[threadIdx.x] = lg;
  __syncthreads();
  for (int o = 128; o > 0; o >>= 1) {
    if (threadIdx.x < o) { r0[threadIdx.x] += r0[threadIdx.x + o]; r1[threadIdx.x] += r1[threadIdx.x + o]; }
    __syncthreads();
  }
  if (threadIdx.x == 0) { atomicAdd(&lamsum[k], r0[0]); atomicAdd(lgl, r1[0]); }
}

// ---------------- gamma (doc conc.) stats: T4 = sum_d [lgamma(sum gam) - sum lgamma(gam)]
__global__ void gam_stats_kernel(const float* __restrict__ dlv, float* __restrict__ T4a) {
  int d = blockIdx.x;
  int k = threadIdx.x;  // 64 threads
  float a = __expf(dlv[(size_t)d * K + k]);
  __shared__ float r0[64], r1[64];
  r0[k] = a; r1[k] = lgammaf(a);
  __syncthreads();
  for (int o = 32; o > 0; o >>= 1) {
    if (k < o) { r0[k] += r0[k + o]; r1[k] += r1[k + o]; }
    __syncthreads();
  }
  if (k == 0) atomicAdd(T4a, lgammaf(r0[0]) - r1[0]);
}

// ---------------- topic gamma draws + row sums ----------------
__global__ void topic_gamma_kernel(const float* __restrict__ tlv,
                                   float* __restrict__ gbuf,
                                   float* __restrict__ rowsum) {
  int v = blockIdx.x * 256 + threadIdx.x;
  int k = blockIdx.y, s = blockIdx.z;
  float g = 0.0f;
  if (v < V) {
    float a = __expf(tlv[(size_t)k * V + v]);
    uint64_t id = ((uint64_t)s * K + k) * V + v;
    g = gamma_sample(a, id);
    gbuf[((size_t)s * K + k) * V + v] = g;
  }
  __shared__ float red[256];
  red[threadIdx.x] = g; __syncthreads();
  for (int o = 128; o > 0; o >>= 1) {
    if (threadIdx.x < o) red[threadIdx.x] += red[threadIdx.x + o];
    __syncthreads();
  }
  if (threadIdx.x == 0) atomicAdd(&rowsum[s * K + k], red[0]);
}

// ---------------- normalize topics + T1 = sum (lam-1) log topic ----------------
__global__ void topic_norm_kernel(const float* __restrict__ tlv,
                                  float* __restrict__ gbuf,
                                  const float* __restrict__ rowsum,
                                  float* __restrict__ T1) {
  int v = blockIdx.x * 256 + threadIdx.x;
  int k = blockIdx.y, s = blockIdx.z;
  float contrib = 0.0f;
  if (v < V) {
    float rs = fmaxf(rowsum[s * K + k], 1e-30f);
    size_t idx = ((size_t)s * K + k) * V + v;
    float t = fmaxf(gbuf[idx] / rs, 1e-30f);
    gbuf[idx] = t;
    float lam = __expf(tlv[(size_t)k * V + v]);
    contrib = (lam - 1.0f) * __logf(t);
  }
  __shared__ float red[256];
  red[threadIdx.x] = contrib; __syncthreads();
  for (int o = 128; o > 0; o >>= 1) {
    if (threadIdx.x < o) red[threadIdx.x] += red[threadIdx.x + o];
    __syncthreads();
  }
  if (threadIdx.x == 0) atomicAdd(&T1[s], red[0]);
}

// ---------------- doc_topics: sample + normalize + T3, one block per (s,d) row ----
__global__ void doc_sample_kernel(const float* __restrict__ dlv,
                                  float* __restrict__ dt,
                                  float* __restrict__ T3) {
  int sd = blockIdx.x;
  int s = sd / D, d = sd % D;
  int k = threadIdx.x;  // 64 threads
  float a = __expf(dlv[(size_t)d * K + k]);
  uint64_t id = (uint64_t)S * K * V + ((uint64_t)s * D + d) * K + k;
  float g = gamma_sample(a, id);
  __shared__ float red[64];
  red[k] = g; __syncthreads();
  for (int o = 32; o > 0; o >>= 1) { if (k < o) red[k] += red[k + o]; __syncthreads(); }
  float denom = fmaxf(red[0], 1e-30f);
  __syncthreads();
  float t = fmaxf(g / denom, 1e-30f);
  dt[((size_t)s * D + d) * K + k] = t;
  red[k] = (a - 1.0f) * __logf(t); __syncthreads();
  for (int o = 32; o > 0; o >>= 1) { if (k < o) red[k] += red[k + o]; __syncthreads(); }
  if (k == 0) atomicAdd(&T3[s], red[0]);
}

// ---------------- fused WMMA GEMM + masked log-likelihood ----------------
// grid: (ceil(V/128), S), block: 256 (8 waves). Each wave: 16x16 C tile via
// 16x v_wmma_f32_16x16x4_f32; block loops over 32 D-tiles of 16.
// B tile staged via GLOBAL_LOAD_ASYNC_TO_LDS_B32; A tiles double-buffered with
// GLOBAL_LOAD_ASYNC_TO_LDS_B128 prefetched one iteration ahead (ASYNCcnt).
__global__ void loglik_wmma_kernel(const float* __restrict__ topics,      // (S,K,V)
                                   const float* __restrict__ doc_topics,  // (S,D,K)
                                   const int*   __restrict__ docs,        // (D,V)
                                   float* __restrict__ loglik) {          // (S)
  __shared__ __align__(16) float Bs[64][132];     // topics tile  K x 128 (+pad)
  __shared__ __align__(16) float As[2][16][68];   // doc tiles    16 x 64 (+pad), dbl-buf
  __shared__ float red[256];

  const int s    = blockIdx.y;
  const int v0   = blockIdx.x * 128;
  const int tid  = threadIdx.x;
  const int wave = tid >> 5;
  const int lane = tid & 31;
  const int half = lane >> 4;
  const int mn   = lane & 15;
  const int n0   = wave * 16;

  const float* Tp = topics + (size_t)s * K * V;
  const float* Ap = doc_topics + (size_t)s * D * K;

  const int jB    = tid & 127;     // B-tile column owned by this thread
  const int kbase = tid >> 7;      // B-tile row base (rows kbase + 2n)
  const int vB    = v0 + jB;
  const int mA    = tid >> 4;            // A-tile row for async b128 copy
  const int kA    = (tid & 15) << 2;     // A-tile col (16B chunk)

#if USE_ASYNC_LDS
  // ---- stage B tile: direct global->LDS async copies (no VGPR round trip) ----
  #pragma unroll
  for (int n = 0; n < 32; ++n) {
    int k = kbase + 2 * n;
    if (vB < V) {
      __builtin_amdgcn_global_load_async_to_lds_b32(
          (AS1 int*)(Tp + (size_t)k * V + vB),
          (AS3 int*)&Bs[k][jB], 0, 0);
    } else {
      Bs[k][jB] = 0.0f;   // zero-fill tail columns so WMMA never sees garbage
    }
  }
  // ---- prefetch first A tile into buffer 0 ----
  __builtin_amdgcn_global_load_async_to_lds_b128(
      (AS1 i32x4*)(Ap + (size_t)mA * K + kA),
      (AS3 i32x4*)&As[0][mA][kA], 0, 0);
#else
  // fallback: batched load via VGPRs
  for (int n = 0; n < 32; n += 8) {
    float tmp[8];
    #pragma unroll
    for (int u = 0; u < 8; ++u) {
      int k = kbase + 2 * (n + u);
      tmp[u] = (vB < V) ? Tp[(size_t)k * V + vB] : 0.0f;
    }
    #pragma unroll
    for (int u = 0; u < 8; ++u) Bs[kbase + 2 * (n + u)][jB] = tmp[u];
  }
#endif

  float acc = 0.0f;

  for (int it = 0; it < D / 16; ++it) {
    const int cur = it & 1;
    const int d0  = it * 16;

#if USE_ASYNC_LDS
    if (it + 1 < D / 16) {
      // prefetch next A tile into the other buffer, then wait for current one
      __builtin_amdgcn_global_load_async_to_lds_b128(
          (AS1 i32x4*)(Ap + (size_t)(d0 + 16 + mA) * K + kA),
          (AS3 i32x4*)&As[cur ^ 1][mA][kA], 0, 0);
      __builtin_amdgcn_s_wait_asynccnt(1);
    } else {
      __builtin_amdgcn_s_wait_asynccnt(0);
    }
    __syncthreads();   // all waves' async LDS writes for As[cur] (and Bs) visible
#else
    __syncthreads();   // previous readers of As[cur] done
    for (int i = tid; i < 16 * 64; i += 256) {
      int m = i >> 6, k2 = i & 63;
      As[cur][m][k2] = Ap[(size_t)(d0 + m) * K + k2];
    }
    __syncthreads();
#endif

    v8f c = {};
    #pragma unroll
    for (int k0 = 0; k0 < 64; k0 += 4) {
      v2f a, b;
      // A 16x4 frag: lane m=mn, VGPRv <- K = k0 + v + 2*half
      a.x = As[cur][mn][k0 + 2 * half + 0];
      a.y = As[cur][mn][k0 + 2 * half + 1];
      // B 4x16 frag: lane n=mn, VGPRv <- K = k0 + v + 2*half
      b.x = Bs[k0 + 2 * half + 0][n0 + mn];
      b.y = Bs[k0 + 2 * half + 1][n0 + mn];
      c = __builtin_amdgcn_wmma_f32_16x16x4_f32(false, a, false, b,
                                                (short)0, c, false, false);
    }

    int v = v0 + n0 + mn;
    if (v < V) {
      #pragma unroll
      for (int r = 0; r < 8; ++r) {
        int d = d0 + r + 8 * half;             // C VGPR r: M = r + 8*half
        int cnt = docs[(size_t)d * V + v];
        if (cnt > 0) acc += (float)cnt * __logf(c[r] + EPS);
      }
    }

#if USE_ASYNC_LDS
    __syncthreads();   // all readers done with As[cur] before it is prefetch-overwritten
#endif
  }

  red[tid] = acc; __syncthreads();
  for (int o = 128; o > 0; o >>= 1) {
    if (tid < o) red[tid] += red[tid + o];
    __syncthreads();
  }
  if (tid == 0) atomicAdd(&loglik[s], red[0]);
}

// ---------------- finalize: assemble -ELBO ----------------
__global__ void finalize_kernel(const float* __restrict__ lamsum,
                                const float* __restrict__ lgl,
                                const float* __restrict__ T4a,
                                const float* __restrict__ T1,
                                const float* __restrict__ T3,
                                const float* __restrict__ LL,
                                float* __restrict__ out) {
  int k = threadIdx.x;  // 64 threads
  __shared__ float red[64];
  red[k] = lgammaf(fmaxf(lamsum[k], 1e-30f));
  __syncthreads();
  for (int o = 32; o > 0; o >>= 1) { if (k < o) red[k] += red[k + o]; __syncthreads(); }
  if (k == 0) {
    float T2 = red[0] - lgl[0];
    float T4 = T4a[0];
    float Kl = (float)K * lgammaf((float)V);   // log_p_topics (eta=1)
    float Dl = (float)D * lgammaf((float)K);   // log_p_doc_topics (alpha=1)
    float elbo = 0.0f;
    for (int s = 0; s < S; ++s)
      elbo += Kl + Dl + LL[s] - (T1[s] + T2 + T3[s] + T4);
    out[0] = -(elbo / (float)S);
  }
}

extern "C" void kernel_launch(void* const* d_in, const int* in_sizes, int n_in,
                              void* d_out, int out_size, void* d_ws, size_t ws_size,
                              hipStream_t stream) {
  (void)in_sizes; (void)n_in; (void)out_size; (void)ws_size;
  const float* tlv  = (const float*)d_in[0];  // (K,V)
  const float* dlv  = (const float*)d_in[1];  // (D,K)
  const int*   docs = (const int*)d_in[2];    // (D,V)
  float* out = (float*)d_out;

  char* ws = (char*)d_ws;
  size_t off = 0;
  float* topics = (float*)(ws + off); off += (size_t)S * K * V * sizeof(float);
  float* dtb    = (float*)(ws + off); off += (size_t)S * D * K * sizeof(float);
  float* acc    = (float*)(ws + off);
  float* rowsum = acc;               // S*K
  float* lamsum = rowsum + S * K;    // K
  float* lgl    = lamsum + K;        // 1
  float* T4a    = lgl + 1;           // 1
  float* T1     = T4a + 1;           // S
  float* T3     = T1 + S;            // S
  float* LL     = T3 + S;            // S
  size_t accN   = (size_t)S * K + K + 2 + 3 * S;

  (void)hipMemsetAsync(acc, 0, accN * sizeof(float), stream);

  dim3 gKV((V + 255) / 256, K, 1);
  lam_stats_kernel<<<gKV, 256, 0, stream>>>(tlv, lamsum, lgl);
  gam_stats_kernel<<<D, 64, 0, stream>>>(dlv, T4a);

  dim3 gSKV((V + 255) / 256, K, S);
  topic_gamma_kernel<<<gSKV, 256, 0, stream>>>(tlv, topics, rowsum);
  topic_norm_kernel<<<gSKV, 256, 0, stream>>>(tlv, topics, rowsum, T1);

  doc_sample_kernel<<<S * D, 64, 0, stream>>>(dlv, dtb, T3);

  dim3 gLL((V + 127) / 128, S, 1);
  loglik_wmma_kernel<<<gLL, 256, 0, stream>>>(topics, dtb, docs, LL);

  finalize_kernel<<<1, 64, 0, stream>>>(lamsum, lgl, T4a, T1, T3, LL, out);
}